// HPBC_steptest_23201413333272
// MI455X (gfx1250) — compile-verified
//
#include <hip/hip_runtime.h>

// ---------------------------------------------------------------------------
// HPBC step for MI455X (gfx1250, wave32).
//
// acc[i,p] = sum_m E[i-m,p] * R_m[i]
// R_m[i]   = sum_n c1[m,n]*Q1_m[i-n] + c2[m,n]*Q2_m[i-n]
// Q1_m[u]  = sum_p E[u,p]*conj(F[u-m,p]),  Q2_m[u] = sum_p F[u,p]*conj(E[u-m,p])
//
// CDNA5 specifics: async global->LDS halo staging (ASYNCcnt +
// s_wait_asynccnt), wave-uniform coefficient reads on the SMEM/constant-cache
// path, interleaved float2 LDS layout for single-ds_load_b64 taps.
// No WMMA: the (i-m, i-n, i-m-n) index coupling admits no GEMM factorization,
// and im2col onto V_WMMA_F32_16X16X4_F32 uses 2/16 columns < packed VALU FMA.
// ---------------------------------------------------------------------------

#define N_SEQ   65536
#define L_TRIM  50
#define NBATCH  8
#define NPOL    2
#define KPAIRS  449
#define TILE    256
#define HALO    50                     // max |m+n|
#define QHALO   25                     // max |n| (and |m|)
#define WIN     (TILE + 2*HALO)        // 356 samples of E,F per block
#define QWIN    (TILE + 2*QHALO)       // 306 samples of Q1,Q2 per block
#define NOUT    (N_SEQ - 2*L_TRIM)     // 65436 trimmed outputs per batch
#define NTILES  ((NOUT + TILE - 1) / TILE)  // 256

// Low 32 bits of a flat shared-space pointer == LDS byte address.
__device__ __forceinline__ unsigned lds_addr(const void* p) {
  return (unsigned)(unsigned long long)p;
}

// CDNA5 async global->LDS copy (GVS mode: SADDR64 + VADDR32), ASYNCcnt.
__device__ __forceinline__ void async_g2l_b64(unsigned lds, const float* base,
                                              unsigned byte_off) {
  asm volatile("global_load_async_to_lds_b64 %0, %1, %2"
               :: "v"(lds), "v"(byte_off), "s"(base) : "memory");
}
__device__ __forceinline__ void wait_async_all() {
  asm volatile("s_wait_asynccnt 0" ::: "memory");
}

__global__ __launch_bounds__(TILE)
void hpbc_step_kernel(const float* __restrict__ Er_g, const float* __restrict__ Ei_g,
                      const float* __restrict__ Fr_g, const float* __restrict__ Fi_g,
                      const float* __restrict__ C1r_g, const float* __restrict__ C1i_g,
                      const float* __restrict__ C2r_g, const float* __restrict__ C2i_g,
                      const float* __restrict__ tinfo,
                      float* __restrict__ out) {
  // E/F halo windows: sample w holds time index (i0 - HALO + w), both pols.
  __shared__ float sEr[WIN][NPOL], sEi[WIN][NPOL];
  __shared__ float sFr[WIN][NPOL], sFi[WIN][NPOL];
  // Per-m pol-summed products (interleaved re/im -> one ds_load_b64 per tap).
  __shared__ float2 sQ1[QWIN], sQ2[QWIN];

  const int t  = (int)threadIdx.x;
  const int b  = (int)blockIdx.y;
  const int i0 = L_TRIM + (int)blockIdx.x * TILE;

  const size_t bbase = (size_t)b * N_SEQ * NPOL;
  const float* Er = Er_g + bbase;
  const float* Ei = Ei_g + bbase;
  const float* Fr = Fr_g + bbase;
  const float* Fi = Fi_g + bbase;

  // ---- choose_C: Fs/2e9 is exact in f32 (values 20/40/80/160) ----
  const int x = __float2int_rn(tinfo[b * 4 + 2] * 5.0e-10f);
  const int ind = (x == 20) ? 0 : (x == 40) ? 1 : (x == 80) ? 2 : (x == 160) ? 3 : 0;
  // scale = sqrt(P)^3, P = 10^(dBm/10) / Nmodes
  const float P = __powf(10.0f, tinfo[b * 4 + 0] * 0.1f) / (float)NPOL;
  const float scale = P * __fsqrt_rn(P);

  // Wave-uniform coefficient rows: k is uniform -> s_load (constant cache).
  const float* __restrict__ C1r = C1r_g + (size_t)ind * KPAIRS;
  const float* __restrict__ C1i = C1i_g + (size_t)ind * KPAIRS;
  const float* __restrict__ C2r = C2r_g + (size_t)ind * KPAIRS;
  const float* __restrict__ C2i = C2i_g + (size_t)ind * KPAIRS;

  // ---- stage E/F halo windows into LDS via the async data mover ----
  for (int w = t; w < WIN; w += TILE) {
    unsigned j = (unsigned)((i0 - HALO + w) & (N_SEQ - 1));  // circular roll
    unsigned off = j * (NPOL * 4u);                          // 8 B = both pols
    async_g2l_b64(lds_addr(&sEr[w][0]), Er, off);
    async_g2l_b64(lds_addr(&sEi[w][0]), Ei, off);
    async_g2l_b64(lds_addr(&sFr[w][0]), Fr, off);
    async_g2l_b64(lds_addr(&sFi[w][0]), Fi, off);
  }
  wait_async_all();
  __syncthreads();

  float a0r = 0.f, a0i = 0.f, a1r = 0.f, a1i = 0.f;  // acc for pol 0 / pol 1
  int k = 0;  // linear (m asc, n asc) pair index — matches Python order

  for (int m = -QHALO; m <= QHALO; ++m) {
    __syncthreads();  // previous iteration's Q consumers are done

    // ---- build Q1_m, Q2_m cooperatively (ds_load_b64 pairs, b64 stores) ----
    for (int u = t; u < QWIN; u += TILE) {
      const int eu = u + (HALO - QHALO);  // window-local index of time u
      const int fu = eu - m;              // window-local index of time u-m
      const float e0r = sEr[eu][0], e0i = sEi[eu][0];
      const float e1r = sEr[eu][1], e1i = sEi[eu][1];
      const float f0r = sFr[fu][0], f0i = sFi[fu][0];
      const float f1r = sFr[fu][1], f1i = sFi[fu][1];
      const float g0r = sFr[eu][0], g0i = sFi[eu][0];
      const float g1r = sFr[eu][1], g1i = sFi[eu][1];
      const float h0r = sEr[fu][0], h0i = sEi[fu][0];
      const float h1r = sEr[fu][1], h1i = sEi[fu][1];
      // Q1 = E[u]·conj(F[u-m]), pol-summed
      sQ1[u] = make_float2(e0r * f0r + e0i * f0i + e1r * f1r + e1i * f1i,
                           e0i * f0r - e0r * f0i + e1i * f1r - e1r * f1i);
      // Q2 = F[u]·conj(E[u-m]), pol-summed
      sQ2[u] = make_float2(g0r * h0r + g0i * h0i + g1r * h1r + g1i * h1i,
                           g0i * h0r - g0r * h0i + g1i * h1r - g1r * h1i);
    }
    __syncthreads();

    // ---- n-convolution: R = sum_n c1·Q1[i-n] + c2·Q2[i-n] ----
    const int mm = (m < 0) ? -m : m;
    const int nmax = (mm == 0) ? QHALO : (QHALO / mm);  // |m*n| <= 25
    float Rr = 0.f, Ri = 0.f;
    for (int n = -nmax; n <= nmax; ++n, ++k) {
      const int q = t + QHALO - n;
      const float2 q1 = sQ1[q];            // one ds_load_b64, conflict-free
      const float2 q2 = sQ2[q];
      const float c1re = C1r[k], c1im = C1i[k];   // uniform -> s_load_b32
      const float c2re = C2r[k], c2im = C2i[k];
      Rr = fmaf(c1re, q1.x, Rr); Rr = fmaf(-c1im, q1.y, Rr);
      Rr = fmaf(c2re, q2.x, Rr); Rr = fmaf(-c2im, q2.y, Rr);
      Ri = fmaf(c1re, q1.y, Ri); Ri = fmaf(c1im, q1.x, Ri);
      Ri = fmaf(c2re, q2.y, Ri); Ri = fmaf(c2im, q2.x, Ri);
    }

    // ---- acc[p] += E[i-m, p] * R ----
    const int em = t + HALO - m;
    const float e0r = sEr[em][0], e0i = sEi[em][0];
    const float e1r = sEr[em][1], e1i = sEi[em][1];
    a0r += e0r * Rr - e0i * Ri;
    a0i += e0r * Ri + e0i * Rr;
    a1r += e1r * Rr - e1i * Ri;
    a1i += e1r * Ri + e1i * Rr;
  }

  // ---- epilogue: E_trim and Eo_trim = F + scale*acc, complex64 layout ----
  const int i = i0 + t;
  if (i < N_SEQ - L_TRIM) {
    const int ec = t + HALO;  // window-local index of time i
    const size_t cbase = ((size_t)b * NOUT + (size_t)(i - L_TRIM)) * NPOL;
    float* oE  = out;
    float* oEo = out + (size_t)NBATCH * NOUT * NPOL * 2;
    // 16-B aligned quads -> global_store_b128 each
    oE[(cbase + 0) * 2 + 0] = sEr[ec][0];
    oE[(cbase + 0) * 2 + 1] = sEi[ec][0];
    oE[(cbase + 1) * 2 + 0] = sEr[ec][1];
    oE[(cbase + 1) * 2 + 1] = sEi[ec][1];
    oEo[(cbase + 0) * 2 + 0] = sFr[ec][0] + scale * a0r;
    oEo[(cbase + 0) * 2 + 1] = sFi[ec][0] + scale * a0i;
    oEo[(cbase + 1) * 2 + 0] = sFr[ec][1] + scale * a1r;
    oEo[(cbase + 1) * 2 + 1] = sFi[ec][1] + scale * a1i;
  }
}

extern "C" void kernel_launch(void* const* d_in, const int* in_sizes, int n_in,
                              void* d_out, int out_size, void* d_ws, size_t ws_size,
                              hipStream_t stream) {
  (void)in_sizes; (void)n_in; (void)out_size; (void)d_ws; (void)ws_size;
  dim3 grid(NTILES, NBATCH);
  hipLaunchKernelGGL(hpbc_step_kernel, grid, dim3(TILE), 0, stream,
                     (const float*)d_in[0], (const float*)d_in[1],
                     (const float*)d_in[2], (const float*)d_in[3],
                     (const float*)d_in[4], (const float*)d_in[5],
                     (const float*)d_in[6], (const float*)d_in[7],
                     (const float*)d_in[8],
                     (float*)d_out);
}